// CodeBook_52175262712108
// MI455X (gfx1250) — compile-verified
//
#include <hip/hip_runtime.h>
#include <hip/hip_bf16.h>
#include <math.h>

// Problem constants (from reference): mlc_emb (8,128,1024) f32, weight (8192,1024) f32
#define BS        8
#define SEQ       128
#define HID       1024
#define NROWS     (BS * SEQ)      // 1024 total rows
#define NCODES    8192            // codebook entries
#define LSAP_N    SEQ             // 128 rows per batch
#define LSAP_M    NCODES          // 8192 cols per batch
#define INF_F     1e30f

typedef __attribute__((ext_vector_type(2))) float v2f;
typedef __attribute__((ext_vector_type(8))) float v8f;

// ---------------------------------------------------------------------------
// Kernel 1: squared row norms.  out[row] = sum_k x[row][k]^2
// ---------------------------------------------------------------------------
__global__ __launch_bounds__(256)
void sqnorm_kernel(const float* __restrict__ x, float* __restrict__ out, int ncols)
{
    int row = blockIdx.x;
    const float* p = x + (size_t)row * ncols;
    float s = 0.0f;
    for (int c = threadIdx.x; c < ncols; c += 256) {
        float t = p[c];
        s += t * t;
    }
    for (int off = 16; off > 0; off >>= 1) s += __shfl_down(s, off);
    __shared__ float red[8];
    if ((threadIdx.x & 31) == 0) red[threadIdx.x >> 5] = s;
    __syncthreads();
    if (threadIdx.x == 0) {
        float t = 0.0f;
        #pragma unroll
        for (int w = 0; w < 8; ++w) t += red[w];
        out[row] = t;
    }
}

// ---------------------------------------------------------------------------
// Kernel 2: WMMA f32 distance GEMM, register-blocked 16x64 per wave.
// dist[m][n] = sqrt(max(rowsq[m] + colsq[n] - 2 * dot(flat[m], weight[n]), 0))
// One A fragment is reused across 4 B fragments / 4 accumulators per K-step,
// cutting A-side fragment traffic 4x and doubling WMMA density in the loop.
//
// A (16x4 f32, 2 VGPR): lanes 0-15 -> M=lane, K={0,1}; lanes 16-31 -> K={2,3}
// B (4x16 f32, 2 VGPR): lanes 0-15 -> N=lane, K={0,1}; lanes 16-31 -> K={2,3}
// C/D (16x16 f32, 8 VGPR): VGPR v, lanes 0-15 -> M=v; lanes 16-31 -> M=v+8
// ---------------------------------------------------------------------------
__global__ __launch_bounds__(256)
void dist_gemm_kernel(const float* __restrict__ A,      // (1024, 1024) row-major
                      const float* __restrict__ W,      // (8192, 1024) row-major
                      const float* __restrict__ rowsq,  // (1024,)
                      const float* __restrict__ colsq,  // (8192,)
                      float* __restrict__ dist)         // (1024, 8192)
{
    const int K = HID;
    int lane = threadIdx.x & 31;
    int wave = threadIdx.x >> 5;
    int tile = blockIdx.x * 8 + wave;       // 8192 supertiles: 64 (M) x 128 (N)
    int tm   = tile >> 7;                   // / 128
    int tn   = tile & 127;
    int m0   = tm << 4;                     // 16 rows
    int n0   = tn << 6;                     // 64 cols

    int half = lane >> 4;                   // 0: K pair {0,1}; 1: K pair {2,3}
    int l16  = lane & 15;

    const float* Ap  = A + (size_t)(m0 + l16) * K + (half << 1);
    const float* Wp0 = W + (size_t)(n0 +  0 + l16) * K + (half << 1);
    const float* Wp1 = W + (size_t)(n0 + 16 + l16) * K + (half << 1);
    const float* Wp2 = W + (size_t)(n0 + 32 + l16) * K + (half << 1);
    const float* Wp3 = W + (size_t)(n0 + 48 + l16) * K + (half << 1);

    v8f acc0 = {}, acc1 = {}, acc2 = {}, acc3 = {};
    for (int k = 0; k < K; k += 4) {
        if ((k & 63) == 0) {
            __builtin_prefetch(Ap  + k + 64, 0, 3);   // WGP-scope prefetch
            __builtin_prefetch(Wp0 + k + 64, 0, 3);
            __builtin_prefetch(Wp1 + k + 64, 0, 3);
            __builtin_prefetch(Wp2 + k + 64, 0, 3);
            __builtin_prefetch(Wp3 + k + 64, 0, 3);
        }
        v2f a  = *(const v2f*)(Ap  + k);
        v2f b0 = *(const v2f*)(Wp0 + k);
        v2f b1 = *(const v2f*)(Wp1 + k);
        v2f b2 = *(const v2f*)(Wp2 + k);
        v2f b3 = *(const v2f*)(Wp3 + k);
        acc0 = __builtin_amdgcn_wmma_f32_16x16x4_f32(false, a, false, b0,
                                                     (short)0, acc0, false, false);
        acc1 = __builtin_amdgcn_wmma_f32_16x16x4_f32(false, a, false, b1,
                                                     (short)0, acc1, false, false);
        acc2 = __builtin_amdgcn_wmma_f32_16x16x4_f32(false, a, false, b2,
                                                     (short)0, acc2, false, false);
        acc3 = __builtin_amdgcn_wmma_f32_16x16x4_f32(false, a, false, b3,
                                                     (short)0, acc3, false, false);
    }

    #pragma unroll
    for (int vi = 0; vi < 8; ++vi) {
        int row  = m0 + vi + (half << 3);
        float rq = rowsq[row];
        int col  = n0 + l16;
        size_t base = (size_t)row * NCODES;
        dist[base + col +  0] = sqrtf(fmaxf(rq + colsq[col +  0] - 2.0f * acc0[vi], 0.0f));
        dist[base + col + 16] = sqrtf(fmaxf(rq + colsq[col + 16] - 2.0f * acc1[vi], 0.0f));
        dist[base + col + 32] = sqrtf(fmaxf(rq + colsq[col + 32] - 2.0f * acc2[vi], 0.0f));
        dist[base + col + 48] = sqrtf(fmaxf(rq + colsq[col + 48] - 2.0f * acc3[vi], 0.0f));
    }
}

// ---------------------------------------------------------------------------
// Kernel 3: Jonker-Volgenant shortest augmenting path LSAP, one block per batch.
// n=128 rows, m=8192 cols. Mirrors the reference solve_lsap exactly, including
// argmin lowest-index tie-break. Big per-batch arrays live in global workspace
// (hot in L2); small state in LDS.  SC == !remaining (invariant of the ref).
// ---------------------------------------------------------------------------
__global__ __launch_bounds__(1024)
void lsap_kernel(const float* __restrict__ dist,        // (1024, 8192)
                 float* __restrict__ v_g,               // (8, 8192)
                 float* __restrict__ spc_g,             // (8, 8192)
                 int*   __restrict__ path_g,            // (8, 8192)
                 int*   __restrict__ row4col_g,         // (8, 8192)
                 unsigned char* __restrict__ rem_g,     // (8, 8192)
                 int*   __restrict__ indices)           // (1024,)
{
    const int b = blockIdx.x;
    const float* D = dist + (size_t)b * LSAP_N * LSAP_M;
    float* v        = v_g      + (size_t)b * LSAP_M;
    float* spc      = spc_g    + (size_t)b * LSAP_M;
    int*   path     = path_g   + (size_t)b * LSAP_M;
    int*   row4col  = row4col_g+ (size_t)b * LSAP_M;
    unsigned char* rem = rem_g + (size_t)b * LSAP_M;

    __shared__ float u_s[LSAP_N];
    __shared__ int   col4row_s[LSAP_N];
    __shared__ unsigned char SR_s[LSAP_N];
    __shared__ float redV[32];
    __shared__ int   redJ[32];
    __shared__ float b_minVal;
    __shared__ int   b_i;
    __shared__ int   b_sink;

    const int tid = threadIdx.x;
    const int nt  = blockDim.x;   // 1024

    if (tid < LSAP_N) { u_s[tid] = 0.0f; col4row_s[tid] = -1; }
    for (int j = tid; j < LSAP_M; j += nt) { v[j] = 0.0f; row4col[j] = -1; }
    __syncthreads();

    for (int cur = 0; cur < LSAP_N; ++cur) {
        // ----- per-row init -----
        if (tid < LSAP_N) SR_s[tid] = 0;
        for (int j = tid; j < LSAP_M; j += nt) {
            spc[j] = INF_F; path[j] = -1; rem[j] = 1;
        }
        if (tid == 0) { b_i = cur; b_minVal = 0.0f; b_sink = -1; }
        __syncthreads();

        // ----- shortest augmenting path -----
        for (int iter = 0; iter < LSAP_M; ++iter) {
            int   i      = b_i;
            float minVal = b_minVal;
            if (tid == 0) SR_s[i] = 1;
            float ui = u_s[i];
            const float* Drow = D + (size_t)i * LSAP_M;

            float lv = INF_F;
            int   lj = -1;
            for (int j = tid; j < LSAP_M; j += nt) {
                if (rem[j]) {
                    float r = minVal + Drow[j] - ui - v[j];
                    float s = spc[j];
                    if (r < s) { s = r; spc[j] = r; path[j] = i; }
                    if (s < lv || (s == lv && (unsigned)j < (unsigned)lj)) {
                        lv = s; lj = j;
                    }
                }
            }
            // wave32 argmin reduce (value, then lowest index on ties)
            for (int off = 16; off > 0; off >>= 1) {
                float ov = __shfl_down(lv, off);
                int   oj = __shfl_down(lj, off);
                if (ov < lv || (ov == lv && (unsigned)oj < (unsigned)lj)) {
                    lv = ov; lj = oj;
                }
            }
            if ((tid & 31) == 0) { redV[tid >> 5] = lv; redJ[tid >> 5] = lj; }
            __syncthreads();
            if (tid < 32) {
                lv = redV[tid]; lj = redJ[tid];
                for (int off = 16; off > 0; off >>= 1) {
                    float ov = __shfl_down(lv, off);
                    int   oj = __shfl_down(lj, off);
                    if (ov < lv || (ov == lv && (unsigned)oj < (unsigned)lj)) {
                        lv = ov; lj = oj;
                    }
                }
                if (tid == 0) {
                    int j = lj;
                    b_minVal = lv;
                    rem[j] = 0;                    // SC[j] = true
                    int rc = row4col[j];
                    if (rc < 0) b_sink = j; else b_i = rc;
                }
            }
            __syncthreads();
            if (b_sink >= 0) break;                 // uniform condition
        }

        // ----- dual update -----
        const int   sink   = b_sink;
        const float minVal = b_minVal;
        if (tid < LSAP_N) {
            if (tid == cur) {
                u_s[tid] += minVal;
            } else if (SR_s[tid]) {
                u_s[tid] += minVal - spc[col4row_s[tid]];
            }
        }
        for (int j = tid; j < LSAP_M; j += nt) {
            if (!rem[j]) v[j] -= minVal - spc[j];   // SC columns
        }
        __syncthreads();

        // ----- augment along stored path (sequential, thread 0) -----
        if (tid == 0) {
            int j = sink;
            for (int a = 0; a <= LSAP_N; ++a) {
                int i = path[j];
                row4col[j] = i;
                int prev = col4row_s[i];
                col4row_s[i] = j;
                if (i == cur) break;
                j = prev;
            }
        }
        __syncthreads();
    }

    if (tid < LSAP_N) indices[b * LSAP_N + tid] = col4row_s[tid];
}

// ---------------------------------------------------------------------------
// Kernel 4: gather + straight-through output + loss partial sums.
// out = f + (q - f); loss_sum += (f - q)^2  (q_latent + commitment = 1.25x)
// ---------------------------------------------------------------------------
__global__ __launch_bounds__(256)
void gather_loss_kernel(const float* __restrict__ flat,
                        const float* __restrict__ W,
                        const int*   __restrict__ idx,
                        float* __restrict__ out,
                        float* __restrict__ acc)
{
    int row = blockIdx.x;                       // 0..1023
    int q   = idx[row];
    const float* f = flat + (size_t)row * HID;
    const float* w = W    + (size_t)q   * HID;
    float s = 0.0f;
    for (int c = threadIdx.x; c < HID; c += 256) {
        float fv = f[c], wv = w[c];
        out[(size_t)row * HID + c] = fv + (wv - fv);  // straight-through fwd
        float e = fv - wv;
        s += e * e;
    }
    for (int off = 16; off > 0; off >>= 1) s += __shfl_down(s, off);
    __shared__ float red[8];
    if ((threadIdx.x & 31) == 0) red[threadIdx.x >> 5] = s;
    __syncthreads();
    if (threadIdx.x == 0) {
        float t = 0.0f;
        #pragma unroll
        for (int w8 = 0; w8 < 8; ++w8) t += red[w8];
        atomicAdd(acc, t);
    }
}

// ---------------------------------------------------------------------------
// Kernel 5: finalize — write loss scalar and indices (as floats) into d_out.
// d_out layout: [0, 1048576) quantized_st | [1048576] loss | [1048577, +1024) idx
// ---------------------------------------------------------------------------
__global__ __launch_bounds__(1024)
void finalize_kernel(const float* __restrict__ acc,
                     const int*   __restrict__ idx,
                     float* __restrict__ out)
{
    int t = threadIdx.x;
    if (t == 0) {
        out[(size_t)NROWS * HID] =
            1.25f * acc[0] / (float)((size_t)NROWS * HID);
    }
    if (t < NROWS) {
        out[(size_t)NROWS * HID + 1 + t] = (float)idx[t];
    }
}

// ---------------------------------------------------------------------------
extern "C" void kernel_launch(void* const* d_in, const int* in_sizes, int n_in,
                              void* d_out, int out_size, void* d_ws, size_t ws_size,
                              hipStream_t stream) {
    const float* mlc    = (const float*)d_in[0];   // (8,128,1024) -> flat (1024,1024)
    const float* weight = (const float*)d_in[1];   // (8192,1024)
    float* out = (float*)d_out;

    // Workspace carve-up
    char* ws = (char*)d_ws;
    float* dist    = (float*)ws;  ws += (size_t)NROWS * NCODES * sizeof(float); // 32 MB
    float* v_g     = (float*)ws;  ws += (size_t)BS * LSAP_M * sizeof(float);
    float* spc_g   = (float*)ws;  ws += (size_t)BS * LSAP_M * sizeof(float);
    int*   path_g  = (int*)ws;    ws += (size_t)BS * LSAP_M * sizeof(int);
    int*   r4c_g   = (int*)ws;    ws += (size_t)BS * LSAP_M * sizeof(int);
    unsigned char* rem_g = (unsigned char*)ws; ws += (size_t)BS * LSAP_M;
    float* rowsq   = (float*)ws;  ws += (size_t)NROWS * sizeof(float);
    float* colsq   = (float*)ws;  ws += (size_t)NCODES * sizeof(float);
    int*   indices = (int*)ws;    ws += (size_t)NROWS * sizeof(int);
    float* acc     = (float*)ws;  ws += sizeof(float);

    // 1) squared norms
    sqnorm_kernel<<<NROWS, 256, 0, stream>>>(mlc, rowsq, HID);
    sqnorm_kernel<<<NCODES, 256, 0, stream>>>(weight, colsq, HID);

    // 2) WMMA distance GEMM: 8192 16x64 wave-tiles, 8 waves/block -> 1024 blocks
    dist_gemm_kernel<<<1024, 256, 0, stream>>>(mlc, weight, rowsq, colsq, dist);

    // 3) LSAP: one block per batch
    lsap_kernel<<<BS, 1024, 0, stream>>>(dist, v_g, spc_g, path_g, r4c_g,
                                         rem_g, indices);

    // 4) gather + loss
    hipMemsetAsync(acc, 0, sizeof(float), stream);
    gather_loss_kernel<<<NROWS, 256, 0, stream>>>(mlc, weight, indices, out, acc);

    // 5) finalize scalar loss + indices
    finalize_kernel<<<1, 1024, 0, stream>>>(acc, indices, out);
}